// PEER_78099685310942
// MI455X (gfx1250) — compile-verified
//
#include <hip/hip_runtime.h>
#include <hip/hip_bf16.h>
#include <math.h>

// ---------------------------------------------------------------------------
// PEER (product-key expert retrieval), fp32 end-to-end.
// GEMMs use CDNA5 V_WMMA_F32_16X16X4_F32 (wave32 matrix pipe, full fp32).
// Each GEMM wave computes a 32x64 C tile: 2 M-tiles x 4 N-tiles, so the 8
// strided B loads per K-step feed 8 WMMAs (2x better B reuse than 16x64).
// ---------------------------------------------------------------------------

typedef __attribute__((ext_vector_type(2))) float v2f;
typedef __attribute__((ext_vector_type(8))) float v8f;

#define N_TOK   2048
#define D_MODEL 512
#define Q_COLS  4096     // 2 * HEADS * DIM_KEY
#define HEADS   8
#define NKEYS   256
#define DKEY    256
#define TOPK    16

// workspace layout (in floats)
#define WS_Q_OFF   0u
#define WS_Q_SZ    (N_TOK * Q_COLS)                  // 8,388,608
#define WS_SIM_OFF (WS_Q_OFF + WS_Q_SZ)
#define WS_SIM_SZ  (2 * N_TOK * HEADS * NKEYS)      // 8,388,608
#define WS_W_OFF   (WS_SIM_OFF + WS_SIM_SZ)
#define WS_W_SZ    (N_TOK * HEADS * TOPK)           // 262,144
#define WS_I_OFF   (WS_W_OFF + WS_W_SZ)             // ints, same count

// ---------------------------------------------------------------------------
// Kernel 1: q = x @ W_q   ([2048,512] x [512,4096] -> [2048,4096])
// One wave -> 32x64 output tile. 8 waves per block. 4096 waves total.
// ---------------------------------------------------------------------------
__global__ __launch_bounds__(256) void PEER_qgemm(const float* __restrict__ x,
                                                  const float* __restrict__ Wq,
                                                  float* __restrict__ q) {
    const int lane = threadIdx.x & 31;
    const int wid  = blockIdx.x * 8 + (threadIdx.x >> 5);     // 4096 waves total
    const int tMg  = wid >> 6;           // 0..63  (32-row groups)
    const int tN   = wid & 63;           // 0..63  (64-col groups)
    const int row0 = tMg * 32;
    const int col0 = tN * 64;

    const int rA   = lane & 15;
    const int kOff = (lane >> 4) << 1;   // lanes 0-15: K{0,1}; lanes 16-31: K{2,3}
    const int cB   = lane & 15;

    v8f c00 = {}, c01 = {}, c02 = {}, c03 = {};
    v8f c10 = {}, c11 = {}, c12 = {}, c13 = {};

    const float* aRow0 = x + (size_t)(row0 + rA) * D_MODEL;
    const float* aRow1 = aRow0 + 16 * D_MODEL;
    for (int kk = 0; kk < D_MODEL; kk += 4) {
        const int ka = kk + kOff;
        v2f a0, a1;
        a0.x = aRow0[ka];  a0.y = aRow0[ka + 1];
        a1.x = aRow1[ka];  a1.y = aRow1[ka + 1];
        const float* bp = Wq + (size_t)ka * Q_COLS + col0 + cB;
        v2f b0, b1, b2, b3;
        b0.x = bp[0];   b0.y = bp[Q_COLS];
        b1.x = bp[16];  b1.y = bp[Q_COLS + 16];
        b2.x = bp[32];  b2.y = bp[Q_COLS + 32];
        b3.x = bp[48];  b3.y = bp[Q_COLS + 48];
        c00 = __builtin_amdgcn_wmma_f32_16x16x4_f32(false, a0, false, b0, (short)0, c00, false, false);
        c01 = __builtin_amdgcn_wmma_f32_16x16x4_f32(false, a0, false, b1, (short)0, c01, false, false);
        c02 = __builtin_amdgcn_wmma_f32_16x16x4_f32(false, a0, false, b2, (short)0, c02, false, false);
        c03 = __builtin_amdgcn_wmma_f32_16x16x4_f32(false, a0, false, b3, (short)0, c03, false, false);
        c10 = __builtin_amdgcn_wmma_f32_16x16x4_f32(false, a1, false, b0, (short)0, c10, false, false);
        c11 = __builtin_amdgcn_wmma_f32_16x16x4_f32(false, a1, false, b1, (short)0, c11, false, false);
        c12 = __builtin_amdgcn_wmma_f32_16x16x4_f32(false, a1, false, b2, (short)0, c12, false, false);
        c13 = __builtin_amdgcn_wmma_f32_16x16x4_f32(false, a1, false, b3, (short)0, c13, false, false);
    }

    const int rBase = row0 + ((lane >> 4) << 3);  // M = r (+8 for upper half-lanes)
    const int cc    = col0 + (lane & 15);
#pragma unroll
    for (int r = 0; r < 8; ++r) {
        float* o0 = q + (size_t)(rBase + r) * Q_COLS + cc;
        o0[0]  = c00[r];
        o0[16] = c01[r];
        o0[32] = c02[r];
        o0[48] = c03[r];
        float* o1 = o0 + 16 * Q_COLS;
        o1[0]  = c10[r];
        o1[16] = c11[r];
        o1[32] = c12[r];
        o1[48] = c13[r];
    }
}

// ---------------------------------------------------------------------------
// Kernel 2: sim[p,n,h,k] = sum_dk q[n, p*2048+h*256+dk] * keys[h,k,p,dk]
// For each (p,h): [2048,256] x [256,256]^T. One wave -> 32x64 tile.
// ---------------------------------------------------------------------------
__global__ __launch_bounds__(256) void PEER_simgemm(const float* __restrict__ q,
                                                    const float* __restrict__ keys,
                                                    float* __restrict__ sim) {
    const int lane = threadIdx.x & 31;
    const int wid  = blockIdx.x * 8 + (threadIdx.x >> 5);  // 4096 waves
    const int ph   = wid >> 8;          // 0..15  (p*8+h)
    const int rem  = wid & 255;
    const int tMg  = rem >> 2;          // 0..63
    const int tN4  = rem & 3;           // 0..3
    const int p    = ph >> 3;
    const int h    = ph & 7;
    const int row0 = tMg * 32;
    const int col0 = tN4 * 64;

    const int rA   = lane & 15;
    const int kOff = (lane >> 4) << 1;
    const int cB   = lane & 15;

    v8f c00 = {}, c01 = {}, c02 = {}, c03 = {};
    v8f c10 = {}, c11 = {}, c12 = {}, c13 = {};

    const float* aRow0 = q + (size_t)(row0 + rA) * Q_COLS + p * 2048 + h * 256;
    const float* aRow1 = aRow0 + 16 * Q_COLS;
    // keys[h,k,p,dk] at h*131072 + k*512 + p*256 + dk
    const float* kbase = keys + (size_t)h * 131072 + p * 256 + (size_t)(col0 + cB) * 512;

    for (int kk = 0; kk < DKEY; kk += 4) {
        const int ka = kk + kOff;
        v2f a0, a1;
        a0.x = aRow0[ka];  a0.y = aRow0[ka + 1];
        a1.x = aRow1[ka];  a1.y = aRow1[ka + 1];
        v2f b0, b1, b2, b3;
        b0.x = kbase[ka];             b0.y = kbase[ka + 1];
        b1.x = kbase[ 8192 + ka];     b1.y = kbase[ 8192 + ka + 1];
        b2.x = kbase[16384 + ka];     b2.y = kbase[16384 + ka + 1];
        b3.x = kbase[24576 + ka];     b3.y = kbase[24576 + ka + 1];
        c00 = __builtin_amdgcn_wmma_f32_16x16x4_f32(false, a0, false, b0, (short)0, c00, false, false);
        c01 = __builtin_amdgcn_wmma_f32_16x16x4_f32(false, a0, false, b1, (short)0, c01, false, false);
        c02 = __builtin_amdgcn_wmma_f32_16x16x4_f32(false, a0, false, b2, (short)0, c02, false, false);
        c03 = __builtin_amdgcn_wmma_f32_16x16x4_f32(false, a0, false, b3, (short)0, c03, false, false);
        c10 = __builtin_amdgcn_wmma_f32_16x16x4_f32(false, a1, false, b0, (short)0, c10, false, false);
        c11 = __builtin_amdgcn_wmma_f32_16x16x4_f32(false, a1, false, b1, (short)0, c11, false, false);
        c12 = __builtin_amdgcn_wmma_f32_16x16x4_f32(false, a1, false, b2, (short)0, c12, false, false);
        c13 = __builtin_amdgcn_wmma_f32_16x16x4_f32(false, a1, false, b3, (short)0, c13, false, false);
    }

    const int rBase = row0 + ((lane >> 4) << 3);
    const int cc    = col0 + (lane & 15);
#pragma unroll
    for (int r = 0; r < 8; ++r) {
        float* o0 = sim + (((size_t)(p * N_TOK + rBase + r) * HEADS + h) * NKEYS) + cc;
        o0[0]  = c00[r];
        o0[16] = c01[r];
        o0[32] = c02[r];
        o0[48] = c03[r];
        float* o1 = o0 + (size_t)16 * HEADS * NKEYS;
        o1[0]  = c10[r];
        o1[16] = c11[r];
        o1[32] = c12[r];
        o1[48] = c13[r];
    }
}

// ---------------------------------------------------------------------------
// Kernel 3: top-16 of sim0, top-16 of sim1, cartesian 16x16, top-16 of 256,
// softmax(scores) -> weights + flat pk_idx. One wave per (n,h).
// ---------------------------------------------------------------------------
__device__ __forceinline__ void wave_top16(float vals[8], int lane,
                                           float* __restrict__ outv,
                                           int* __restrict__ outi) {
    for (int t = 0; t < 16; ++t) {
        float bv = -3.402823466e38f;
        int   bi = 0x7fffffff;
#pragma unroll
        for (int j = 0; j < 8; ++j) {
            const int idx = lane + (j << 5);
            if (vals[j] > bv) { bv = vals[j]; bi = idx; }   // strict > keeps lowest idx in-lane
        }
#pragma unroll
        for (int off = 16; off > 0; off >>= 1) {
            const float ov = __shfl_xor(bv, off, 32);
            const int   oi = __shfl_xor(bi, off, 32);
            if (ov > bv || (ov == bv && oi < bi)) { bv = ov; bi = oi; }
        }
        if (lane == 0) { outv[t] = bv; if (outi) outi[t] = bi; }
        if ((bi & 31) == lane) vals[bi >> 5] = -3.402823466e38f;  // knock out winner
    }
}

__global__ __launch_bounds__(256) void PEER_topk(const float* __restrict__ sim,
                                                 float* __restrict__ wout,
                                                 int* __restrict__ iout) {
    __shared__ float s0[8][16];
    __shared__ float s1[8][16];
    __shared__ float sv[8][16];
    __shared__ int   si[8][16];

    const int lane = threadIdx.x & 31;
    const int w    = threadIdx.x >> 5;
    const int task = blockIdx.x * 8 + w;      // 16384 tasks
    const int n    = task >> 3;
    const int h    = task & 7;

    const float* r0 = sim + ((size_t)n * HEADS + h) * NKEYS;                 // p=0
    const float* r1 = sim + ((size_t)(N_TOK + n) * HEADS + h) * NKEYS;       // p=1

    float v0[8], v1[8];
#pragma unroll
    for (int j = 0; j < 8; ++j) {
        v0[j] = r0[lane + (j << 5)];
        v1[j] = r1[lane + (j << 5)];
    }
    wave_top16(v0, lane, s0[w], nullptr);
    wave_top16(v1, lane, s1[w], nullptr);
    // intra-wave LDS RAW is in-order (DScnt ordering) -> safe to read now
    float cv[8];
#pragma unroll
    for (int j = 0; j < 8; ++j) {
        const int c = lane + (j << 5);        // flat cartesian index 0..255
        cv[j] = s0[w][c >> 4] + s1[w][c & 15];
    }
    wave_top16(cv, lane, sv[w], si[w]);

    // softmax over the 16 selected scores (uniform across lanes)
    float m = -3.402823466e38f;
#pragma unroll
    for (int t = 0; t < 16; ++t) m = fmaxf(m, sv[w][t]);
    float s = 0.0f;
#pragma unroll
    for (int t = 0; t < 16; ++t) s = s + __expf(sv[w][t] - m);
    const float inv = 1.0f / s;

    if (lane < 16) {
        wout[(task << 4) + lane] = __expf(sv[w][lane] - m) * inv;
        iout[(task << 4) + lane] = si[w][lane];
    }
}

// ---------------------------------------------------------------------------
// Kernel 4: expert pass. One block (8 waves = 8 heads) per token.
// out[n,:] = sum_{h,t} w[n,h,t] * gelu(x[n]·down[idx]) * up[idx]
// ---------------------------------------------------------------------------
__global__ __launch_bounds__(256) void PEER_experts(const float* __restrict__ x,
                                                    const float* __restrict__ down,
                                                    const float* __restrict__ up,
                                                    const float* __restrict__ wsw,
                                                    const int* __restrict__ wsi,
                                                    float* __restrict__ out) {
    __shared__ float xs[D_MODEL];
    __shared__ float part[HEADS][D_MODEL];

    const int n    = blockIdx.x;
    const int tid  = threadIdx.x;
    const int lane = tid & 31;
    const int h    = tid >> 5;

    xs[tid]       = x[(size_t)n * D_MODEL + tid];
    xs[tid + 256] = x[(size_t)n * D_MODEL + tid + 256];
    __syncthreads();

    float acc[16];
#pragma unroll
    for (int j = 0; j < 16; ++j) acc[j] = 0.0f;

    const int base = ((n * HEADS) + h) * TOPK;
    for (int t = 0; t < TOPK; ++t) {
        const int   idx = wsi[base + t];
        const float wgt = wsw[base + t];
        const float* dr = down + (size_t)idx * D_MODEL;
        float p = 0.0f;
#pragma unroll
        for (int j = 0; j < 16; ++j) p += xs[lane + (j << 5)] * dr[lane + (j << 5)];
#pragma unroll
        for (int off = 16; off > 0; off >>= 1) p += __shfl_xor(p, off, 32);
        const float g  = 0.5f * p * (1.0f + erff(p * 0.70710678118654752440f));
        const float cw = wgt * g;
        const float* ur = up + (size_t)idx * D_MODEL;
#pragma unroll
        for (int j = 0; j < 16; ++j) acc[j] += cw * ur[lane + (j << 5)];
    }

#pragma unroll
    for (int j = 0; j < 16; ++j) part[h][lane + (j << 5)] = acc[j];
    __syncthreads();

    for (int d = tid; d < D_MODEL; d += 256) {
        float s = 0.0f;
#pragma unroll
        for (int hh = 0; hh < HEADS; ++hh) s += part[hh][d];
        out[(size_t)n * D_MODEL + d] = s;
    }
}

// ---------------------------------------------------------------------------
extern "C" void kernel_launch(void* const* d_in, const int* in_sizes, int n_in,
                              void* d_out, int out_size, void* d_ws, size_t ws_size,
                              hipStream_t stream) {
    const float* x     = (const float*)d_in[0];   // [1,2048,512]
    const float* Wq    = (const float*)d_in[1];   // [512,4096]
    const float* keys  = (const float*)d_in[2];   // [8,256,2,256]
    const float* down  = (const float*)d_in[3];   // [65536,512]
    const float* up    = (const float*)d_in[4];   // [65536,512]
    float* out = (float*)d_out;                   // [1,2048,512]

    float* ws   = (float*)d_ws;
    float* q    = ws + WS_Q_OFF;
    float* sim  = ws + WS_SIM_OFF;
    float* wsw  = ws + WS_W_OFF;
    int*   wsi  = (int*)(ws + WS_I_OFF);

    // K1: q = x @ W_q   (64*64 = 4096 waves -> 512 blocks of 8 waves)
    PEER_qgemm<<<512, 256, 0, stream>>>(x, Wq, q);
    // K2: sim = q . keys (16 * 64 * 4 = 4096 waves -> 512 blocks)
    PEER_simgemm<<<512, 256, 0, stream>>>(q, keys, sim);
    // K3: top-k + softmax (16384 wave-tasks -> 2048 blocks of 8 waves)
    PEER_topk<<<2048, 256, 0, stream>>>(sim, wsw, wsi);
    // K4: expert aggregation (one block per token)
    PEER_experts<<<2048, 256, 0, stream>>>(x, down, up, wsw, wsi, out);
}